// MedianConv_3178275799594
// MI455X (gfx1250) — compile-verified
//
#include <hip/hip_runtime.h>

typedef __attribute__((ext_vector_type(2))) float v2f;
typedef __attribute__((ext_vector_type(8))) float v8f;

#define FIN  128
#define FOUT 128
#define KNBR 32

// ---------------------------------------------------------------------------
// Kernel 1: h[N,FOUT] = x[N,FIN] @ W[FIN,FOUT] via V_WMMA_F32_16X16X4_F32.
// 8 waves per block, one 16x16 output tile per wave, K accumulated in steps
// of 4 (32 chained WMMAs). All-lane-active blocks => EXEC all 1s for WMMA.
// (GEMM is only 1.6 GFLOP / ~51 MB of HBM traffic: microseconds; f32 WMMA
// keeps full precision so downstream median *comparisons* match reference.)
// ---------------------------------------------------------------------------
__global__ void __launch_bounds__(256)
medconv_gemm_wmma(const float* __restrict__ x, const float* __restrict__ W,
                  float* __restrict__ h, int n_rows) {
  const int lane = threadIdx.x & 31;
  const int wave = threadIdx.x >> 5;
  const int tile = blockIdx.x * 8 + wave;
  const int tiles_n = FOUT / 16;            // 8 col tiles
  const int tm = tile / tiles_n;
  const int tn = tile % tiles_n;
  if (tm * 16 >= n_rows) return;            // full waves only (N % 16 == 0)

  // A layout: lane m = lane&15 (rows), K pair base = 2*(lane>>4)
  const int m  = lane & 15;
  const int kb = (lane >> 4) << 1;          // 0 or 2
  const float* __restrict__ xrow = x + (size_t)(tm * 16 + m) * FIN;
  // B layout: lane n = lane&15 (cols), same K pair base (row index of W)
  const float* __restrict__ wcol = W + tn * 16 + (lane & 15);

  v8f c = {};
  for (int k0 = 0; k0 < FIN; k0 += 4) {
    v2f a, b;
    a.x = xrow[k0 + kb + 0];
    a.y = xrow[k0 + kb + 1];
    b.x = wcol[(size_t)(k0 + kb + 0) * FOUT];
    b.y = wcol[(size_t)(k0 + kb + 1) * FOUT];
    // (neg_a, A, neg_b, B, c_mod, C, reuse_a, reuse_b)
    c = __builtin_amdgcn_wmma_f32_16x16x4_f32(false, a, false, b,
                                              (short)0, c, false, false);
  }

  // C/D layout: row = tm*16 + r + 8*(lane>>4), col = tn*16 + (lane&15)
  const int half = lane >> 4;
  float* __restrict__ out =
      h + (size_t)(tm * 16 + half * 8) * FOUT + tn * 16 + (lane & 15);
#pragma unroll
  for (int r = 0; r < 8; ++r) out[(size_t)r * FOUT] = c[r];
}

// ---------------------------------------------------------------------------
// Batcher odd-even mergesort network, fully unrolled => all constant indices,
// array lives in VGPRs. N=16 -> 63 compare-exchanges (vs 80 bitonic).
// ---------------------------------------------------------------------------
template <int NN>
__device__ __forceinline__ void oemsort(float* v) {
#pragma unroll
  for (int p = 1; p < NN; p <<= 1) {
#pragma unroll
    for (int k = p; k >= 1; k >>= 1) {
#pragma unroll
      for (int j = k % p; j + k < NN; j += 2 * k) {
#pragma unroll
        for (int i = 0; i < k; ++i) {
          if (i + j + k < NN) {
            if ((i + j) / (2 * p) == (i + j + k) / (2 * p)) {
              const float a = v[i + j], b = v[i + j + k];
              v[i + j]     = fminf(a, b);
              v[i + j + k] = fmaxf(a, b);
            }
          }
        }
      }
    }
  }
}

// ---------------------------------------------------------------------------
// Kernel 2: per node n, per feature f: lower median (sorted[15]) of
// h[nbrs[n][0..31]][f].
//
// One wave per node, each lane owns 4 features: every neighbor row is ONE
// global_load_b128 (512B = the full row) out of L2 (h fits easily in the
// 192MB L2, so the 819MB gather never touches HBM). Median per feature via
// two Batcher-sorted 16-element halves (2x63 CEs) + the two-sorted-array
// order-statistic identity rank15 = max_i min(a[i], b[15-i]) (31 ops),
// ~40% fewer VALU ops than a full bitonic-32 sort.
// ---------------------------------------------------------------------------
__global__ void __launch_bounds__(256)
medconv_gather_median(const float* __restrict__ h, const int* __restrict__ nbrs,
                      float* __restrict__ out, int n_nodes) {
  __shared__ int snbr[8][KNBR];
  const int lane = threadIdx.x & 31;
  const int wave = threadIdx.x >> 5;
  const int n = blockIdx.x * 8 + wave;
  if (n >= n_nodes) return;                 // 50000 % 8 == 0: never taken

  // Stage this wave's 32 neighbor indices (one per lane) in LDS; only this
  // wave reads them back, LDS ops are in-order per wave.
  snbr[wave][lane] = nbrs[(size_t)n * KNBR + lane];

  float va[4][16], vb[4][16];               // [component][neighbor-half]
  const float* __restrict__ hf = h + (size_t)lane * 4;
#pragma unroll
  for (int j = 0; j < 16; ++j) {
    const float4 g = *(const float4*)(hf + (size_t)snbr[wave][j] * FOUT);
    va[0][j] = g.x; va[1][j] = g.y; va[2][j] = g.z; va[3][j] = g.w;
  }
#pragma unroll
  for (int j = 0; j < 16; ++j) {
    const float4 g = *(const float4*)(hf + (size_t)snbr[wave][j + 16] * FOUT);
    vb[0][j] = g.x; vb[1][j] = g.y; vb[2][j] = g.z; vb[3][j] = g.w;
  }

  float r[4];
#pragma unroll
  for (int comp = 0; comp < 4; ++comp) {
    oemsort<16>(va[comp]);
    oemsort<16>(vb[comp]);
    // 16th smallest (0-indexed rank 15) of the union of two sorted 16-arrays
    float med = fminf(va[comp][0], vb[comp][15]);
#pragma unroll
    for (int i = 1; i < 16; ++i)
      med = fmaxf(med, fminf(va[comp][i], vb[comp][15 - i]));
    r[comp] = med;
  }

  float4 res;
  res.x = r[0]; res.y = r[1]; res.z = r[2]; res.w = r[3];
  *(float4*)(out + (size_t)n * FOUT + (size_t)lane * 4) = res;
}

// ---------------------------------------------------------------------------
extern "C" void kernel_launch(void* const* d_in, const int* in_sizes, int n_in,
                              void* d_out, int out_size, void* d_ws, size_t ws_size,
                              hipStream_t stream) {
  const float* x    = (const float*)d_in[0];   // [N, FIN] f32
  const int*   nbrs = (const int*)  d_in[1];   // [N, K] i32
  const float* W    = (const float*)d_in[2];   // [FIN, FOUT] f32
  float*       out  = (float*)d_out;           // [N, FOUT] f32
  float*       h    = (float*)d_ws;            // [N, FOUT] f32 scratch (25.6 MB)

  const int N = in_sizes[0] / FIN;             // 50000
  const int row_tiles = (N + 15) / 16;         // 3125 (exact)
  const int tiles = row_tiles * (FOUT / 16);   // 25000
  const int gemm_blocks = (tiles + 7) / 8;     // 8 waves/block, 1 tile/wave

  medconv_gemm_wmma<<<gemm_blocks, 256, 0, stream>>>(x, W, h, N);

  const int med_blocks = (N + 7) / 8;          // 8 waves/block, 1 node/wave
  medconv_gather_median<<<med_blocks, 256, 0, stream>>>(h, nbrs, out, N);
}